// GATModel_51848845197729
// MI455X (gfx1250) — compile-verified
//
#include <hip/hip_runtime.h>
#include <math.h>

typedef __attribute__((ext_vector_type(2))) float v2f;
typedef __attribute__((ext_vector_type(8))) float v8f;

#define NEG_SLOPE 0.2f
#define BN_EPS 1e-5f

__device__ __forceinline__ float leaky(float x) { return x >= 0.f ? x : NEG_SLOPE * x; }

__device__ __forceinline__ void atomicMaxF(float* addr, float val) {
    unsigned int* ai = (unsigned int*)addr;
    unsigned int old = __float_as_uint(*addr);
    while (__uint_as_float(old) < val) {
        unsigned int assumed = old;
        old = atomicCAS(ai, assumed, __float_as_uint(val));
        if (old == assumed) break;
    }
}

__global__ void k_zero(float* p, int n) {
    int i = blockIdx.x * blockDim.x + threadIdx.x;
    if (i < n) p[i] = 0.f;
}

// edge gaussian weights + incoming-edge count + incoming-weight sum (for self-loop attr)
__global__ void k_edgew(const float* __restrict__ raw, const int* __restrict__ src,
                        const int* __restrict__ dst, float* __restrict__ ew,
                        float* cnt, float* lsum, int E) {
    int e = blockIdx.x * blockDim.x + threadIdx.x;
    if (e >= E) return;
    int s0 = src[e], d0 = dst[e];
    float dx = raw[s0 * 3 + 0] - raw[d0 * 3 + 0];
    float dy = raw[s0 * 3 + 1] - raw[d0 * 3 + 1];
    float dz = raw[s0 * 3 + 2] - raw[d0 * 3 + 2];
    float w = __expf(-(dx * dx + dy * dy + dz * dz) * 2.0f);  // 1/(2*sigma^2)=2, sigma=0.5
    ew[e] = w;
    atomicAdd(&cnt[d0], 1.f);
    atomicAdd(&lsum[d0], w);
}

__global__ void k_loopdiv(float* la, const float* cnt, int N) {
    int i = blockIdx.x * blockDim.x + threadIdx.x;
    if (i < N) la[i] = la[i] / fmaxf(cnt[i], 1.f);
}

// ce[h] = sum_d We[h*D+d] * att_e[h*D+d]   (edge_attr is 1-channel -> per-head scalar)
__global__ void k_ce(const float* We, const float* ae, float* ce, int H, int D) {
    int h = threadIdx.x;
    if (h >= H) return;
    float acc = 0.f;
    for (int d = 0; d < D; ++d) acc += We[h * D + d] * ae[h * D + d];
    ce[h] = acc;
}

// H = X(N x K) @ W(K x 64), WMMA f32 16x16x4, one wave per 16x16 tile.
// blockDim = 128 (4 waves -> 4 column tiles covering all 64 cols), grid = nrow/16.
__global__ void k_gemm_wmma(const float* __restrict__ X, const float* __restrict__ W,
                            float* __restrict__ H, int K) {
    int wave = threadIdx.x >> 5;     // column tile 0..3
    int lane = threadIdx.x & 31;
    int tileM = blockIdx.x;
    int row = tileM * 16 + (lane & 15);
    int col = wave * 16 + (lane & 15);
    int koff = (lane >> 4) * 2;      // lanes 16-31 carry K+2,K+3
    v8f c = {};
    for (int k0 = 0; k0 < K; k0 += 4) {
        v2f a, b;
        const float* xp = X + row * K + k0 + koff;
        a.x = xp[0];
        a.y = xp[1];
        const float* wp = W + (k0 + koff) * 64 + col;
        b.x = wp[0];
        b.y = wp[64];
        c = __builtin_amdgcn_wmma_f32_16x16x4_f32(false, a, false, b, (short)0, c,
                                                  false, false);
    }
    // C layout: VGPR i -> row i (lanes 0-15) / row i+8 (lanes 16-31), col = lane&15
    int orow = tileM * 16 + ((lane >> 4) << 3);
    for (int i = 0; i < 8; ++i) H[(orow + i) * 64 + col] = c[i];
}

// scalar tail for rows not covered by 16-row tiles (not used when N%16==0)
__global__ void k_gemm_tail(const float* X, const float* W, float* H,
                            int startRow, int nRows, int K) {
    int i = blockIdx.x * blockDim.x + threadIdx.x;
    if (i >= nRows * 64) return;
    int r = startRow + (i >> 6), c = i & 63;
    float acc = 0.f;
    for (int k = 0; k < K; ++k) acc += X[r * K + k] * W[k * 64 + c];
    H[r * 64 + c] = acc;
}

// per (node, head): attention dots and self-loop logit -> init segment max m
__global__ void k_attprep(const float* __restrict__ hmat, const float* __restrict__ asrc,
                          const float* __restrict__ adst, const float* __restrict__ la,
                          const float* __restrict__ ce, float* als, float* ald, float* m,
                          int N, int H, int D) {
    int i = blockIdx.x * blockDim.x + threadIdx.x;
    if (i >= N * H) return;
    int n = i / H, h = i % H;
    const float* hp = hmat + n * 64 + h * D;
    float s = 0.f, d = 0.f;
    for (int k = 0; k < D; ++k) {
        s += hp[k] * asrc[h * D + k];
        d += hp[k] * adst[h * D + k];
    }
    als[i] = s;
    ald[i] = d;
    m[i] = leaky(s + d + la[n] * ce[h]);  // self-loop logit seeds the max
}

__global__ void k_edgemax(const int* __restrict__ src, const int* __restrict__ dst,
                          const float* __restrict__ ew, const float* __restrict__ als,
                          const float* __restrict__ ald, const float* __restrict__ ce,
                          float* m, int E, int H) {
    int i = blockIdx.x * blockDim.x + threadIdx.x;
    if (i >= E * H) return;
    int e = i / H, h = i - e * H;
    int s0 = src[e], d0 = dst[e];
    float l = leaky(als[s0 * H + h] + ald[d0 * H + h] + ew[e] * ce[h]);
    atomicMaxF(&m[d0 * H + h], l);
}

// s[n,h] = exp(self_logit - m)   (initializes softmax denominator with self loop)
__global__ void k_sinit(const float* als, const float* ald, const float* la,
                        const float* ce, const float* m, float* s, int N, int H) {
    int i = blockIdx.x * blockDim.x + threadIdx.x;
    if (i >= N * H) return;
    int n = i / H, h = i - n * H;
    float self = leaky(als[i] + ald[i] + la[n] * ce[h]);
    s[i] = __expf(self - m[i]);
}

__global__ void k_edgesum(const int* __restrict__ src, const int* __restrict__ dst,
                          const float* __restrict__ ew, const float* __restrict__ als,
                          const float* __restrict__ ald, const float* __restrict__ ce,
                          const float* __restrict__ m, float* s, int E, int H) {
    int i = blockIdx.x * blockDim.x + threadIdx.x;
    if (i >= E * H) return;
    int e = i / H, h = i - e * H;
    int s0 = src[e], d0 = dst[e];
    float l = leaky(als[s0 * H + h] + ald[d0 * H + h] + ew[e] * ce[h]);
    atomicAdd(&s[d0 * H + h], __expf(l - m[d0 * H + h]));
}

// out[n,c] = alpha_self * h[n,c]
__global__ void k_selfagg(const float* __restrict__ hmat, const float* __restrict__ als,
                          const float* __restrict__ ald, const float* __restrict__ la,
                          const float* __restrict__ ce, const float* __restrict__ m,
                          const float* __restrict__ s, float* outb, int N, int H, int lgD) {
    int i = blockIdx.x * blockDim.x + threadIdx.x;
    if (i >= N * 64) return;
    int n = i >> 6, c = i & 63, h = c >> lgD;
    int nh = n * H + h;
    float self = leaky(als[nh] + ald[nh] + la[n] * ce[h]);
    float a = __expf(self - m[nh]) / s[nh];
    outb[i] = a * hmat[i];
}

// out[dst,c] += alpha(e,h) * h[src,c]  (64 lanes per edge)
__global__ void k_edgeagg(const int* __restrict__ src, const int* __restrict__ dst,
                          const float* __restrict__ ew, const float* __restrict__ als,
                          const float* __restrict__ ald, const float* __restrict__ ce,
                          const float* __restrict__ m, const float* __restrict__ s,
                          const float* __restrict__ hmat, float* outb, int E, int H, int lgD) {
    int i = blockIdx.x * blockDim.x + threadIdx.x;
    if (i >= E * 64) return;
    int e = i >> 6, c = i & 63, h = c >> lgD;
    int s0 = src[e], d0 = dst[e];
    int dh = d0 * H + h;
    float l = leaky(als[s0 * H + h] + ald[dh] + ew[e] * ce[h]);
    float a = __expf(l - m[dh]) / s[dh];
    atomicAdd(&outb[d0 * 64 + c], a * hmat[s0 * 64 + c]);
}

__global__ void k_bnelu(const float* __restrict__ outb, const float* __restrict__ bias,
                        const float* __restrict__ gamma, const float* __restrict__ beta,
                        const float* __restrict__ rm, const float* __restrict__ rv,
                        float* xout, int N) {
    int i = blockIdx.x * blockDim.x + threadIdx.x;
    if (i >= N * 64) return;
    int c = i & 63;
    float v = outb[i] + bias[c];
    v = (v - rm[c]) * (gamma[c] * rsqrtf(rv[c] + BN_EPS)) + beta[c];
    xout[i] = v > 0.f ? v : __expf(v) - 1.f;
}

__global__ void k_pool(const float* __restrict__ x2, const int* __restrict__ batch,
                       float* pooled, float* cntB, int N) {
    int i = blockIdx.x * blockDim.x + threadIdx.x;
    if (i >= N * 64) return;
    int n = i >> 6, c = i & 63;
    int b = batch[n];
    atomicAdd(&pooled[b * 64 + c], x2[i]);
    if (c == 0) atomicAdd(&cntB[b], 1.f);
}

__global__ void k_fc(const float* __restrict__ pooled, const float* __restrict__ cntB,
                     const float* __restrict__ Wc, const float* __restrict__ bc,
                     float* out, int B) {
    int t = blockIdx.x * blockDim.x + threadIdx.x;
    if (t >= B * 10) return;
    int b = t / 10, k = t - b * 10;
    float inv = 1.f / fmaxf(cntB[b], 1.f);
    float acc = bc[k];
    for (int c = 0; c < 64; ++c) acc += pooled[b * 64 + c] * inv * Wc[c * 10 + k];
    out[t] = acc;
}

static inline int gdiv(int n, int b) { return (n + b - 1) / b; }

extern "C" void kernel_launch(void* const* d_in, const int* in_sizes, int n_in,
                              void* d_out, int out_size, void* d_ws, size_t ws_size,
                              hipStream_t stream) {
    const float* x     = (const float*)d_in[0];
    const float* raw   = (const float*)d_in[1];
    const int*   ei    = (const int*)d_in[2];
    const int*   batch = (const int*)d_in[3];
    const float* W1    = (const float*)d_in[4];
    const float* asrc1 = (const float*)d_in[5];
    const float* adst1 = (const float*)d_in[6];
    const float* We1   = (const float*)d_in[7];
    const float* ae1   = (const float*)d_in[8];
    const float* b1    = (const float*)d_in[9];
    const float* g1    = (const float*)d_in[10];
    const float* be1   = (const float*)d_in[11];
    const float* rm1   = (const float*)d_in[12];
    const float* rv1   = (const float*)d_in[13];
    const float* W2    = (const float*)d_in[14];
    const float* asrc2 = (const float*)d_in[15];
    const float* adst2 = (const float*)d_in[16];
    const float* We2   = (const float*)d_in[17];
    const float* ae2   = (const float*)d_in[18];
    const float* b2    = (const float*)d_in[19];
    const float* g2    = (const float*)d_in[20];
    const float* be2   = (const float*)d_in[21];
    const float* rm2   = (const float*)d_in[22];
    const float* rv2   = (const float*)d_in[23];
    const float* Wc    = (const float*)d_in[24];
    const float* bc    = (const float*)d_in[25];
    float* out = (float*)d_out;

    const int N = in_sizes[3];
    const int E = in_sizes[2] / 2;
    const int B = out_size / 10;
    const int IN_CH = in_sizes[0] / N;
    const int* src = ei;
    const int* dst = ei + E;

    // workspace layout (floats)
    float* ws = (float*)d_ws;
    size_t o = 0;
    float* ew     = ws + o; o += (size_t)E;
    float* cnt    = ws + o; o += (size_t)N;
    float* la     = ws + o; o += (size_t)N;       // loop attr: sum then mean
    float* hmat   = ws + o; o += (size_t)N * 64;
    float* x1     = ws + o; o += (size_t)N * 64;  // conv1 output / conv2 input / conv2 output
    float* als    = ws + o; o += (size_t)N * 8;
    float* ald    = ws + o; o += (size_t)N * 8;
    float* m      = ws + o; o += (size_t)N * 8;
    float* s      = ws + o; o += (size_t)N * 8;
    float* outb   = ws + o; o += (size_t)N * 64;
    float* ce     = ws + o; o += 8;
    float* pooled = ws + o; o += (size_t)B * 64;
    float* cntB   = ws + o; o += (size_t)B;
    (void)ws_size; (void)n_in;

    const int T = 256;
    int tiles = N / 16, rem = N % 16;

    // ---- edge weights + self-loop attrs ----
    k_zero<<<gdiv(N, T), T, 0, stream>>>(cnt, N);
    k_zero<<<gdiv(N, T), T, 0, stream>>>(la, N);
    k_zero<<<gdiv(B * 65, T), T, 0, stream>>>(pooled, B * 65);  // pooled + cntB contiguous
    k_edgew<<<gdiv(E, T), T, 0, stream>>>(raw, src, dst, ew, cnt, la, E);
    k_loopdiv<<<gdiv(N, T), T, 0, stream>>>(la, cnt, N);

    // ---- conv1: H=8, D=8, K=IN_CH ----
    {
        const int H = 8, D = 8, lgD = 3;
        k_ce<<<1, H, 0, stream>>>(We1, ae1, ce, H, D);
        k_gemm_wmma<<<tiles, 128, 0, stream>>>(x, W1, hmat, IN_CH);
        if (rem) k_gemm_tail<<<gdiv(rem * 64, T), T, 0, stream>>>(x, W1, hmat, tiles * 16, rem, IN_CH);
        k_attprep<<<gdiv(N * H, T), T, 0, stream>>>(hmat, asrc1, adst1, la, ce, als, ald, m, N, H, D);
        k_edgemax<<<gdiv(E * H, T), T, 0, stream>>>(src, dst, ew, als, ald, ce, m, E, H);
        k_sinit<<<gdiv(N * H, T), T, 0, stream>>>(als, ald, la, ce, m, s, N, H);
        k_edgesum<<<gdiv(E * H, T), T, 0, stream>>>(src, dst, ew, als, ald, ce, m, s, E, H);
        k_selfagg<<<gdiv(N * 64, T), T, 0, stream>>>(hmat, als, ald, la, ce, m, s, outb, N, H, lgD);
        k_edgeagg<<<gdiv(E * 64, T), T, 0, stream>>>(src, dst, ew, als, ald, ce, m, s, hmat, outb, E, H, lgD);
        k_bnelu<<<gdiv(N * 64, T), T, 0, stream>>>(outb, b1, g1, be1, rm1, rv1, x1, N);
    }

    // ---- conv2: H=1, D=64, K=64, input x1, output back into x1 ----
    {
        const int H = 1, D = 64, lgD = 6;
        k_ce<<<1, H, 0, stream>>>(We2, ae2, ce, H, D);
        k_gemm_wmma<<<tiles, 128, 0, stream>>>(x1, W2, hmat, 64);
        if (rem) k_gemm_tail<<<gdiv(rem * 64, T), T, 0, stream>>>(x1, W2, hmat, tiles * 16, rem, 64);
        k_attprep<<<gdiv(N * H, T), T, 0, stream>>>(hmat, asrc2, adst2, la, ce, als, ald, m, N, H, D);
        k_edgemax<<<gdiv(E * H, T), T, 0, stream>>>(src, dst, ew, als, ald, ce, m, E, H);
        k_sinit<<<gdiv(N * H, T), T, 0, stream>>>(als, ald, la, ce, m, s, N, H);
        k_edgesum<<<gdiv(E * H, T), T, 0, stream>>>(src, dst, ew, als, ald, ce, m, s, E, H);
        k_selfagg<<<gdiv(N * 64, T), T, 0, stream>>>(hmat, als, ald, la, ce, m, s, outb, N, H, lgD);
        k_edgeagg<<<gdiv(E * 64, T), T, 0, stream>>>(src, dst, ew, als, ald, ce, m, s, hmat, outb, E, H, lgD);
        k_bnelu<<<gdiv(N * 64, T), T, 0, stream>>>(outb, b2, g2, be2, rm2, rv2, x1, N);
    }

    // ---- global mean pool + classifier ----
    k_pool<<<gdiv(N * 64, T), T, 0, stream>>>(x1, batch, pooled, cntB, N);
    k_fc<<<gdiv(B * 10, T), T, 0, stream>>>(pooled, cntB, Wc, bc, out, B);
}